// LSTM_14121852469249
// MI455X (gfx1250) — compile-verified
//
#include <hip/hip_runtime.h>

typedef float v2f __attribute__((ext_vector_type(2)));
typedef float v8f __attribute__((ext_vector_type(8)));

#define T_STEPS 512
#define D_IN    256
#define H_SZ    512
#define C_SZ    128
#define N_B     64
#define NWG     64
#define BLK     128            /* 4 waves: one per gate */
#define FC_BLK  256
#define NH      (N_B * H_SZ)   /* 32768 */

// ---------------- WMMA helpers (fp32 16x16x4, fp32 accumulate) ----------------

__device__ __forceinline__ v8f wmma_f32_k4(v2f a, v2f b, v8f c) {
  // 8-arg pattern: (neg_a, A, neg_b, B, c_mod, C, reuse_a, reuse_b)
  return __builtin_amdgcn_wmma_f32_16x16x4_f32(false, a, false, b, (short)0, c,
                                               false, false);
}

// Dual-M-tile K loop: one B fetch feeds two independent WMMA chains.
// ap0/ap1 point at A[m_row]*K + 2*lanehalf for the two 16-row M tiles;
// bp points at W[j_row]*K + 2*lanehalf.
__device__ __forceinline__ void gemm_k2(const float* __restrict__ ap0,
                                        const float* __restrict__ ap1,
                                        const float* __restrict__ bp,
                                        int K, v8f& acc0, v8f& acc1) {
#pragma unroll 4
  for (int k = 0; k < K; k += 4) {
    v2f b  = *(const v2f*)(bp + k);
    v2f a0 = *(const v2f*)(ap0 + k);
    v2f a1 = *(const v2f*)(ap1 + k);
    acc0 = wmma_f32_k4(a0, b, acc0);
    acc1 = wmma_f32_k4(a1, b, acc1);
  }
}

// Single-tile variant (used by the FC epilogue).
__device__ __forceinline__ v8f gemm_k(const float* __restrict__ ap,
                                      const float* __restrict__ bp,
                                      int K, v8f acc) {
#pragma unroll 4
  for (int k = 0; k < K; k += 4) {
    v2f a = *(const v2f*)(ap + k);
    v2f b = *(const v2f*)(bp + k);
    acc = wmma_f32_k4(a, b, acc);
  }
  return acc;
}

// ---------------- device-scope grid barrier (persistent kernel) ----------------

__device__ __forceinline__ void grid_barrier(unsigned* cnt, unsigned target) {
  __builtin_amdgcn_fence(__ATOMIC_RELEASE, "agent");
  __syncthreads();
  if (threadIdx.x == 0) {
    __hip_atomic_fetch_add(cnt, 1u, __ATOMIC_RELEASE, __HIP_MEMORY_SCOPE_AGENT);
    while (__hip_atomic_load(cnt, __ATOMIC_ACQUIRE, __HIP_MEMORY_SCOPE_AGENT) < target)
      __builtin_amdgcn_s_sleep(1);
  }
  __syncthreads();
  __builtin_amdgcn_fence(__ATOMIC_ACQUIRE, "agent");
}

__device__ __forceinline__ float sigm(float v) { return 1.0f / (1.0f + expf(-v)); }

// ---------------- persistent 2-layer LSTM ----------------
// Grid: 64 WGs x 128 threads (4 waves). WG g owns h-columns [(g>>1)*16, +16)
// and batch rows [(g&1)*32, +32). Wave w = gate w; each wave computes BOTH
// 16-row M tiles of its batch half (B reused, 2 independent WMMA chains).
// hA/hB: ping-pong [2][64][512] buffers in workspace (L2 resident).
// Cell states live in LDS for the whole run; written to d_out at the end.

__global__ void __launch_bounds__(BLK)
lstm_persistent(const float* __restrict__ x,    // [512,64,256]
                const float* __restrict__ c0,   // [2,64,512]
                const float* __restrict__ W0,   // [2048,256]
                const float* __restrict__ U0,   // [2048,512]
                const float* __restrict__ b0,   // [2048]
                const float* __restrict__ W1,   // [2048,512]
                const float* __restrict__ U1,   // [2048,512]
                const float* __restrict__ b1,   // [2048]
                float* __restrict__ hA,         // ws [2][64][512]
                float* __restrict__ hB,         // ws [2][64][512]
                unsigned* __restrict__ bar,
                float* __restrict__ cA_out,     // d_out c_n[0]
                float* __restrict__ cB_out)     // d_out c_n[1]
{
  __shared__ float gS[4][32][17];   // gate staging, padded vs bank conflicts
  __shared__ float cAs[32][16];     // layer-0 cell state slice (persistent)
  __shared__ float cBs[32][16];     // layer-1 cell state slice (persistent)

  const int g    = blockIdx.x;
  const int cc   = g >> 1;                 // h-column chunk (16 cols)
  const int mh   = g & 1;                  // batch half (32 rows)
  const int tid  = threadIdx.x;
  const int lane = tid & 31;
  const int gate = tid >> 5;               // 0=i 1=f 2=o 3=c~
  const int l16  = lane & 15;
  const int lh   = lane >> 4;              // lane half -> K offset 2*lh
  const int ar0  = mh * 32 + l16;          // A row, M tile 0
  const int ar1  = ar0 + 16;               // A row, M tile 1
  const int jrow = gate * H_SZ + cc * 16 + l16;  // this lane's weight row

  // load initial cell state slices
  for (int e = tid; e < 32 * 16; e += BLK) {
    int ml = e >> 4, col = e & 15;
    int gi = (mh * 32 + ml) * H_SZ + cc * 16 + col;
    cAs[ml][col] = c0[gi];
    cBs[ml][col] = c0[NH + gi];
  }
  __syncthreads();

  unsigned epoch = 0;

  for (int t = 0; t < T_STEPS; ++t) {
    const int cur = t & 1, nxt = cur ^ 1;
    const float* xt = x + (size_t)t * (N_B * D_IN);

    if (t + 1 < T_STEPS)  // uniform branch; hint next timestep's x into cache
      __builtin_prefetch(x + (size_t)(t + 1) * (N_B * D_IN) + tid * 128, 0, 1);

    // ---------------- layer 0: g0 = x_t @ W0^T + hA @ U0^T ----------------
    v8f acc0 = {}, acc1 = {};
    gemm_k2(xt + ar0 * D_IN + 2 * lh, xt + ar1 * D_IN + 2 * lh,
            W0 + (size_t)jrow * D_IN + 2 * lh, D_IN, acc0, acc1);
    {
      const float* ha = hA + cur * NH;
      gemm_k2(ha + ar0 * H_SZ + 2 * lh, ha + ar1 * H_SZ + 2 * lh,
              U0 + (size_t)jrow * H_SZ + 2 * lh, H_SZ, acc0, acc1);
    }
    {
      const int mlb = lh * 8;
#pragma unroll
      for (int r = 0; r < 8; ++r) {
        gS[gate][mlb + r][l16]      = acc0[r];
        gS[gate][16 + mlb + r][l16] = acc1[r];
      }
    }
    __syncthreads();
    {
      float* hAn = hA + nxt * NH;
      for (int e = tid; e < 32 * 16; e += BLK) {
        int ml = e >> 4, col = e & 15;
        int j  = cc * 16 + col;
        float iv = sigm(gS[0][ml][col] + b0[j]);
        float fv = sigm(gS[1][ml][col] + b0[H_SZ + j]);
        float ov = sigm(gS[2][ml][col] + b0[2 * H_SZ + j]);
        float ct = tanhf(gS[3][ml][col] + b0[3 * H_SZ + j]);
        float cn = fv * cAs[ml][col] + iv * ct;
        cAs[ml][col] = cn;
        hAn[(mh * 32 + ml) * H_SZ + j] = ov * tanhf(cn);
      }
    }
    ++epoch;
    grid_barrier(bar, NWG * epoch);

    // ---------------- layer 1: g1 = hA_new @ W1^T + hB @ U1^T ----------------
    acc0 = (v8f){};
    acc1 = (v8f){};
    {
      const float* ha = hA + nxt * NH;
      gemm_k2(ha + ar0 * H_SZ + 2 * lh, ha + ar1 * H_SZ + 2 * lh,
              W1 + (size_t)jrow * H_SZ + 2 * lh, H_SZ, acc0, acc1);
    }
    {
      const float* hb = hB + cur * NH;
      gemm_k2(hb + ar0 * H_SZ + 2 * lh, hb + ar1 * H_SZ + 2 * lh,
              U1 + (size_t)jrow * H_SZ + 2 * lh, H_SZ, acc0, acc1);
    }
    {
      const int mlb = lh * 8;
#pragma unroll
      for (int r = 0; r < 8; ++r) {
        gS[gate][mlb + r][l16]      = acc0[r];
        gS[gate][16 + mlb + r][l16] = acc1[r];
      }
    }
    __syncthreads();
    {
      float* hBn = hB + nxt * NH;
      for (int e = tid; e < 32 * 16; e += BLK) {
        int ml = e >> 4, col = e & 15;
        int j  = cc * 16 + col;
        float iv = sigm(gS[0][ml][col] + b1[j]);
        float fv = sigm(gS[1][ml][col] + b1[H_SZ + j]);
        float ov = sigm(gS[2][ml][col] + b1[2 * H_SZ + j]);
        float ct = tanhf(gS[3][ml][col] + b1[3 * H_SZ + j]);
        float cn = fv * cBs[ml][col] + iv * ct;
        cBs[ml][col] = cn;
        hBn[(mh * 32 + ml) * H_SZ + j] = ov * tanhf(cn);
      }
    }
    ++epoch;
    grid_barrier(bar, NWG * epoch);
  }

  // final cell states -> outputs (each thread wrote these LDS slots itself)
  for (int e = tid; e < 32 * 16; e += BLK) {
    int ml = e >> 4, col = e & 15;
    int gi = (mh * 32 + ml) * H_SZ + cc * 16 + col;
    cA_out[gi] = cAs[ml][col];
    cB_out[gi] = cBs[ml][col];
  }
}

// ---------------- final FC: out = h_b_last @ fc_w^T + fc_b (WMMA) ----------------
// 4 blocks x 8 waves = 32 waves = 4 Mtiles x 8 Ntiles of the 64x128 output.

__global__ void __launch_bounds__(FC_BLK)
fc_out_kernel(const float* __restrict__ hlast,  // [64,512]
              const float* __restrict__ fcw,    // [128,512]
              const float* __restrict__ fcb,    // [128]
              float* __restrict__ out)          // [64,128]
{
  const int tid  = threadIdx.x;
  const int lane = tid & 31, l16 = lane & 15, lh = lane >> 4;
  const int wid  = blockIdx.x * 8 + (tid >> 5);  // 0..31
  const int mt   = wid & 3, nt = wid >> 2;

  v8f acc = {};
  acc = gemm_k(hlast + (mt * 16 + l16) * H_SZ + 2 * lh,
               fcw + (size_t)(nt * 16 + l16) * H_SZ + 2 * lh, H_SZ, acc);
  const float bias = fcb[nt * 16 + l16];
#pragma unroll
  for (int r = 0; r < 8; ++r)
    out[(mt * 16 + lh * 8 + r) * C_SZ + nt * 16 + l16] = acc[r] + bias;
}

// ---------------- h_n copy ----------------

__global__ void __launch_bounds__(FC_BLK)
copy_h_kernel(const float* __restrict__ hAl,
              const float* __restrict__ hBl,
              float* __restrict__ dst) {
  int i = blockIdx.x * blockDim.x + threadIdx.x;  // exactly 32768 threads
  dst[i] = hAl[i];
  dst[NH + i] = hBl[i];
}

// ---------------- launch ----------------

extern "C" void kernel_launch(void* const* d_in, const int* in_sizes, int n_in,
                              void* d_out, int out_size, void* d_ws, size_t ws_size,
                              hipStream_t stream) {
  const float* x   = (const float*)d_in[0];
  const float* h0  = (const float*)d_in[1];
  const float* c0  = (const float*)d_in[2];
  const float* W0  = (const float*)d_in[3];
  const float* U0  = (const float*)d_in[4];
  const float* b0  = (const float*)d_in[5];
  const float* W1  = (const float*)d_in[6];
  const float* U1  = (const float*)d_in[7];
  const float* b1  = (const float*)d_in[8];
  const float* fcw = (const float*)d_in[9];
  const float* fcb = (const float*)d_in[10];

  float* out = (float*)d_out;            // [64*128]
  float* hn  = out + 64 * C_SZ;          // [2*64*512]
  float* cn  = hn + 2 * NH;              // [2*64*512]

  float*    hA  = (float*)d_ws;          // [2][64][512] ping-pong
  float*    hB  = hA + 2 * NH;
  unsigned* bar = (unsigned*)(hB + 2 * NH);

  hipMemsetAsync(bar, 0, 64, stream);
  // initial hidden states into ping buffer 0 (t even reads buf0, writes buf1;
  // after 512 steps the final state lands back in buffer 0)
  hipMemcpyAsync(hA, h0, NH * sizeof(float), hipMemcpyDeviceToDevice, stream);
  hipMemcpyAsync(hB, h0 + NH, NH * sizeof(float), hipMemcpyDeviceToDevice, stream);

  lstm_persistent<<<NWG, BLK, 0, stream>>>(x, c0, W0, U0, b0, W1, U1, b1,
                                           hA, hB, bar, cn, cn + NH);
  copy_h_kernel<<<NH / FC_BLK, FC_BLK, 0, stream>>>(hA, hB, hn);
  fc_out_kernel<<<4, FC_BLK, 0, stream>>>(hB, fcw, fcb, out);
}